// Shashimal6_39865886441855
// MI455X (gfx1250) — compile-verified
//
#include <hip/hip_runtime.h>

typedef __attribute__((ext_vector_type(2))) float v2f;
typedef __attribute__((ext_vector_type(8))) float v8f;

namespace {
constexpr int HW    = 224;
constexpr int TILE  = 16;
constexpr int TPD   = HW / TILE;      // 14 tiles per dimension
constexpr int TPI   = TPD * TPD;      // 196 tiles per image
constexpr int WAVES = 4;              // waves per block
constexpr int BLKX  = TPI / WAVES;    // 49 blocks in x -> 49*4 == 196 exact
}

// Per wave: one 16x16 output tile of one batch image.
//   num[m,n] = dy_m*gy + dx_n*gx   == A_num(16x4) x B_num(4x16)  (K=2 used, K=2,3 zero)
//   r2 [m,n] = dy_m^2 + dx_n^2     == A_r2 (16x4) x B_r2 (4x16)
// A layout (32-bit 16x4): lanes 0-15 hold M=lane, vgpr0=K0, vgpr1=K1; lanes 16-31 hold K2,K3 (zeroed).
// C/D layout: vgpr v, lanes 0-15 -> (M=v, N=lane); lanes 16-31 -> (M=v+8, N=lane-16).
__global__ __launch_bounds__(WAVES * 32) void gaze_cone_wmma(
    const float* __restrict__ xy,
    const float* __restrict__ head,
    float* __restrict__ out)
{
    const int b    = blockIdx.y;
    const int wave = threadIdx.x >> 5;
    const int lane = threadIdx.x & 31;
    const int tile = blockIdx.x * WAVES + wave;     // uniform per wave, < 196 always
    const int row0 = (tile / TPD) * TILE;
    const int col0 = (tile % TPD) * TILE;

    // Block-uniform per-batch scalars (SALU path).
    const float gx = xy[2 * b + 0];
    const float gy = xy[2 * b + 1];
    const float hx = head[2 * b + 0];
    const float hy = head[2 * b + 1];
    const float g2   = gx * gx + gy * gy;
    const float invg = (g2 > 0.0f) ? __frsqrt_rn(g2) : 0.0f;  // denom==0 path -> cos=0 -> mask 0

    const bool  lo  = lane < 16;
    const int   l15 = lane & 15;
    const float dy  = (float)(row0 + l15) - hy;   // row delta for M = l15
    const float dx  = (float)(col0 + l15) - hx;   // col delta for N = l15

    v2f a_num, b_num, a_r2, b_r2;
    a_num.x = lo ? dy        : 0.0f;  a_num.y = lo ? 1.0f      : 0.0f;
    b_num.x = lo ? gy        : 0.0f;  b_num.y = lo ? gx * dx   : 0.0f;
    a_r2.x  = lo ? dy * dy   : 0.0f;  a_r2.y  = lo ? 1.0f      : 0.0f;
    b_r2.x  = lo ? 1.0f      : 0.0f;  b_r2.y  = lo ? dx * dx   : 0.0f;

    v8f zero = {};
    // 8 args: (neg_a, A, neg_b, B, c_mod, C, reuse_a, reuse_b)
    v8f num = __builtin_amdgcn_wmma_f32_16x16x4_f32(false, a_num, false, b_num,
                                                    (short)0, zero, false, false);
    v8f r2  = __builtin_amdgcn_wmma_f32_16x16x4_f32(false, a_r2,  false, b_r2,
                                                    (short)0, zero, false, false);

    const int mbase = lo ? 0 : 8;
    float* dst = out + ((size_t)b * HW + (size_t)(row0 + mbase)) * HW + (col0 + l15);

    #pragma unroll
    for (int v = 0; v < 8; ++v) {
        float r2v  = r2[v];
        float invr = (r2v > 0.0f) ? __frsqrt_rn(r2v) : 0.0f;  // head pixel -> cos=0 -> mask 0
        float c    = num[v] * invr * invg;
        // mask nonzero only for theta < pi/12 (c > 0.966): small-angle acos expansion,
        // theta = sqrt(2u)*(1 + u/12 + 3u^2/160 + 5u^3/896), u = 1-c, exact where it matters;
        // elsewhere the final clamp zeroes the output. u clamped >=0 so c>1 rounding can't NaN.
        // 12/pi is pre-folded into the polynomial: q = (12/pi)*(1 + u/12 + ...).
        float u    = fmaxf(1.0f - c, 0.0f);
        float s    = __builtin_amdgcn_sqrtf(u + u);   // raw v_sqrt_f32, no denorm-rescue path
        float q    = 3.8197186f + u * (3.1830988e-1f + u * (7.1619724e-2f + u * 2.1317261e-2f));
        float mask = fmaxf(1.0f - s * q, 0.0f);       // 1 - (12/pi)*theta, clamped
        dst[(size_t)v * HW] = mask;
    }
}

extern "C" void kernel_launch(void* const* d_in, const int* in_sizes, int n_in,
                              void* d_out, int out_size, void* d_ws, size_t ws_size,
                              hipStream_t stream)
{
    const float* xy   = (const float*)d_in[0];   // [B,2] gaze direction
    const float* head = (const float*)d_in[1];   // [B,2] head point
    float* out = (float*)d_out;                  // [B,1,224,224]
    const int B = in_sizes[0] / 2;               // 1024

    dim3 grid(BLKX, B, 1);
    dim3 block(WAVES * 32, 1, 1);
    hipLaunchKernelGGL(gaze_cone_wmma, grid, block, 0, stream, xy, head, out);
}